// HLCLConv_supervised_90555090468876
// MI455X (gfx1250) — compile-verified
//
#include <hip/hip_runtime.h>
#include <hip/hip_bf16.h>
#include <math.h>

typedef __attribute__((ext_vector_type(2))) float v2f;
typedef __attribute__((ext_vector_type(8))) float v8f;

#if defined(__gfx1250__) && __has_builtin(__builtin_amdgcn_wmma_f32_16x16x4_f32)
#define USE_WMMA_F32 1
#endif

// ---------------------------------------------------------------------------
// Degree / normalization kernels
// ---------------------------------------------------------------------------
__global__ void init_deg_kernel(float* __restrict__ deg, int n) {
    int i = blockIdx.x * blockDim.x + threadIdx.x;
    if (i < n) deg[i] = 1.0f;                 // self-loop contributes 1
}

__global__ void deg_accum_kernel(const int* __restrict__ cols,
                                 float* __restrict__ deg, int e) {
    int i = blockIdx.x * blockDim.x + threadIdx.x;
    if (i < e) atomicAdd(&deg[cols[i]], 1.0f);
}

__global__ void finish_dinv_kernel(float* __restrict__ deg_dinv, int n) {
    int i = blockIdx.x * blockDim.x + threadIdx.x;
    if (i < n) deg_dinv[i] = rsqrtf(deg_dinv[i]);   // deg >= 1 always
}

// ---------------------------------------------------------------------------
// WMMA f32 GEMM:  C[M x Nout] = A[M x K] * W^T   (W is [Nout x K] row-major)
// One block handles a 16-row stripe; wave w handles column tile [16w, 16w+16).
// blockDim.x must be 2*Nout (i.e. 32 * Nout/16).  K, Nout multiples of 16/4.
// EPI: 0 = none, 1 = ELU (alpha=1).  bias may be nullptr.
// ---------------------------------------------------------------------------
template <int EPI>
__global__ void gemm_wmma_f32_kernel(const float* __restrict__ A,
                                     const float* __restrict__ W,
                                     const float* __restrict__ bias,
                                     float* __restrict__ C,
                                     int K, int Nout) {
    const int mBase = blockIdx.x * 16;
    const int wave  = threadIdx.x >> 5;
    const int lane  = threadIdx.x & 31;
    const int nBase = wave * 16;
    const int lrow  = lane & 15;   // A: M-in-tile index | B: N-in-tile index
    const int khalf = lane >> 4;   // selects K pair {0,1} vs {2,3}

    const float* __restrict__ Arow = A + (size_t)(mBase + lrow) * K;
    const float* __restrict__ Wrow = W + (size_t)(nBase + lrow) * K;

    v8f c = {};
#if defined(USE_WMMA_F32)
    #pragma unroll 8
    for (int k0 = 0; k0 < K; k0 += 4) {
        const int ka = k0 + 2 * khalf;
        v2f a, b;
        a.x = Arow[ka];     a.y = Arow[ka + 1];
        b.x = Wrow[ka];     b.y = Wrow[ka + 1];
        // D = A(16x4) * B(4x16) + C ; emits v_wmma_f32_16x16x4_f32
        c = __builtin_amdgcn_wmma_f32_16x16x4_f32(
                /*neg_a=*/false, a, /*neg_b=*/false, b,
                /*c_mod=*/(short)0, c, /*reuse_a=*/false, /*reuse_b=*/false);
    }
#else
    // Scalar fallback matching the WMMA C layout (host-pass parse safety).
    for (int j = 0; j < 8; ++j) {
        const float* ar = A + (size_t)(mBase + j + 8 * khalf) * K;
        const float* wr = W + (size_t)(nBase + lrow) * K;
        float s = 0.f;
        for (int k = 0; k < K; ++k) s = fmaf(ar[k], wr[k], s);
        c[j] = s;
    }
#endif

    // C layout: VGPR j -> row (j + 8*khalf), col lrow within the tile.
    const int cn = nBase + lrow;
    const float bv = bias ? bias[cn] : 0.0f;
    #pragma unroll
    for (int j = 0; j < 8; ++j) {
        const int rm = mBase + j + 8 * khalf;
        float v = c[j] + bv;
        if (EPI == 1) v = (v > 0.0f) ? v : expm1f(v);   // ELU, alpha = 1
        C[(size_t)rm * Nout + cn] = v;
    }
}

// ---------------------------------------------------------------------------
// Aggregation: out[i,:] = dinv[i]^2 * t[i,:]        (self-loop, also inits)
//              out[c,:] += dinv[r]*dinv[c]*t[r,:]   (edge scatter, atomics)
// ---------------------------------------------------------------------------
__global__ void self_init_kernel(const float* __restrict__ t,
                                 const float* __restrict__ dinv,
                                 float* __restrict__ out, long long total) {
    long long i = (long long)blockIdx.x * blockDim.x + threadIdx.x;
    if (i < total) {
        float d = dinv[i >> 7];                 // D_H == 128
        out[i] = d * d * t[i];
    }
}

__global__ void scatter_edges_kernel(const int* __restrict__ rows,
                                     const int* __restrict__ cols,
                                     const float* __restrict__ dinv,
                                     const float* __restrict__ t,
                                     float* __restrict__ out, int E) {
    const int lane  = threadIdx.x & 31;
    int       wave  = (blockIdx.x * blockDim.x + threadIdx.x) >> 5;
    const int nwave = (gridDim.x * blockDim.x) >> 5;
    for (int e = wave; e < E; e += nwave) {
        const int r = rows[e];
        const int c = cols[e];
        const float w = dinv[r] * dinv[c];
        // 32 lanes x float4 = 512B row (global_load_b128 per lane)
        float4 v = ((const float4*)(t + (size_t)r * 128))[lane];
        float* dst = out + (size_t)c * 128 + lane * 4;
        atomicAdd(dst + 0, w * v.x);
        atomicAdd(dst + 1, w * v.y);
        atomicAdd(dst + 2, w * v.z);
        atomicAdd(dst + 3, w * v.w);
    }
}

__global__ void relu_bias_kernel(float* __restrict__ z,
                                 const float* __restrict__ b, long long total) {
    long long i = (long long)blockIdx.x * blockDim.x + threadIdx.x;
    if (i < total) {
        float v = z[i] + b[i & 127];
        z[i] = v > 0.0f ? v : 0.0f;
    }
}

__global__ void add_bias_kernel(float* __restrict__ z,
                                const float* __restrict__ b, long long total) {
    long long i = (long long)blockIdx.x * blockDim.x + threadIdx.x;
    if (i < total) z[i] += b[i & 127];
}

// ---------------------------------------------------------------------------
// Row-wise log-softmax over 64 logits: one wave per row, 2 values per lane.
// ---------------------------------------------------------------------------
__global__ void log_softmax64_kernel(const float* __restrict__ o,
                                     float* __restrict__ res, int N) {
    const int lane = threadIdx.x & 31;
    const int row  = (blockIdx.x * blockDim.x + threadIdx.x) >> 5;
    if (row >= N) return;
    const float* p = o + (size_t)row * 64;
    float a = p[lane];
    float b = p[lane + 32];
    float m = fmaxf(a, b);
    #pragma unroll
    for (int off = 16; off > 0; off >>= 1) m = fmaxf(m, __shfl_xor(m, off, 32));
    float s = __expf(a - m) + __expf(b - m);
    #pragma unroll
    for (int off = 16; off > 0; off >>= 1) s += __shfl_xor(s, off, 32);
    const float lse = m + __logf(s);
    float* q = res + (size_t)row * 64;
    q[lane]      = a - lse;
    q[lane + 32] = b - lse;
}

// ---------------------------------------------------------------------------
// Host-side orchestration
// ---------------------------------------------------------------------------
extern "C" void kernel_launch(void* const* d_in, const int* in_sizes, int n_in,
                              void* d_out, int out_size, void* d_ws, size_t ws_size,
                              hipStream_t stream) {
    const float* x    = (const float*)d_in[0];
    const int*   ei   = (const int*)  d_in[1];
    const float* W1   = (const float*)d_in[2];
    const float* b1   = (const float*)d_in[3];
    const float* W2   = (const float*)d_in[4];
    const float* b2   = (const float*)d_in[5];
    const float* fcW1 = (const float*)d_in[6];
    const float* fcb1 = (const float*)d_in[7];
    const float* fcW2 = (const float*)d_in[8];
    const float* fcb2 = (const float*)d_in[9];

    const int N = in_sizes[0] / 128;   // D_IN = 128
    const int E = in_sizes[1] / 2;     // edge_index is [2, E]
    const int* rows = ei;              // edge_index[0] = sources
    const int* cols = ei + E;          // edge_index[1] = targets

    float* out = (float*)d_out;
    float* zs  = out;                       // [N,128] first output
    float* res = out + (size_t)N * 128;     // [N, 64] second output

    float* ws   = (float*)d_ws;
    float* dinv = ws;                                   // N floats (deg->dinv)
    float* bufA = ws + (((size_t)N + 255) & ~(size_t)255);
    float* bufB = bufA + (size_t)N * 128;

    const long long NF  = (long long)N * 128;
    const int PT   = 256;
    const int gN   = (N + PT - 1) / PT;
    const int gE   = (E + PT - 1) / PT;
    const int gNF  = (int)((NF + PT - 1) / PT);
    const int gMM  = N / 16;           // 16-row stripes (N = 100000 = 6250*16)
    const int gSC  = 4096;             // grid-stride scatter (32768 waves)

    // 1) degrees + symmetric normalization
    init_deg_kernel   <<<gN, PT, 0, stream>>>(dinv, N);
    deg_accum_kernel  <<<gE, PT, 0, stream>>>(cols, dinv, E);
    finish_dinv_kernel<<<gN, PT, 0, stream>>>(dinv, N);

    // 2) layer 1: t1 = x @ W1^T  -> aggregate -> relu(+b1)
    gemm_wmma_f32_kernel<0><<<gMM, 256, 0, stream>>>(x, W1, nullptr, bufA, 128, 128);
    self_init_kernel     <<<gNF, PT, 0, stream>>>(bufA, dinv, bufB, NF);
    scatter_edges_kernel <<<gSC, PT, 0, stream>>>(rows, cols, dinv, bufA, bufB, E);
    relu_bias_kernel     <<<gNF, PT, 0, stream>>>(bufB, b1, NF);

    // 3) layer 2: t2 = z1 @ W2^T -> aggregate into zs (d_out) -> +b2
    gemm_wmma_f32_kernel<0><<<gMM, 256, 0, stream>>>(bufB, W2, nullptr, bufA, 128, 128);
    self_init_kernel     <<<gNF, PT, 0, stream>>>(bufA, dinv, zs, NF);
    scatter_edges_kernel <<<gSC, PT, 0, stream>>>(rows, cols, dinv, bufA, zs, E);
    add_bias_kernel      <<<gNF, PT, 0, stream>>>(zs, b2, NF);

    // 4) head: h = elu(zs @ fcW1^T + fcb1); o = h @ fcW2^T + fcb2; log_softmax
    gemm_wmma_f32_kernel<1><<<gMM, 256, 0, stream>>>(zs, fcW1, fcb1, bufB, 128, 128);
    gemm_wmma_f32_kernel<0><<<gMM, 128, 0, stream>>>(bufB, fcW2, fcb2, bufA, 128, 64);
    log_softmax64_kernel <<<(N + 7) / 8, 256, 0, stream>>>(bufA, res, N);
}